// GraphAttentionBackbone_65481071402674
// MI455X (gfx1250) — compile-verified
//
#include <hip/hip_runtime.h>
#include <hip/hip_bf16.h>

// out[n, h*128+o] = (1/128) * sum_f x[n,f] * W[h,o,f]
// (softmax of a row-constant score matrix is exactly 1/128 per element, so
//  the adjacency input is mathematically dead and never read.)
//
// X: (8192, 256) f32 row-major
// W: (4, 128, 256) f32 -> flat row-major (512, 256) "W_all"
// out: (8192, 512) f32 row-major
//
// GEMM via V_WMMA_F32_16X16X4_F32: one wave per 16x16 output tile,
// K=256 consumed 8-at-a-time (two WMMA per pair of b128 loads).

typedef __attribute__((ext_vector_type(2))) float v2f;
typedef __attribute__((ext_vector_type(4))) float v4f;
typedef __attribute__((ext_vector_type(8))) float v8f;

#define IN_F   256
#define OUT_ALL 512   // HEADS * OUT_F
#define N_ROWS 8192

__global__ __launch_bounds__(256) void gat_gemm_wmma_f32(
    const float* __restrict__ x,     // (8192, 256)
    const float* __restrict__ w,     // (512, 256)  == W_all
    float* __restrict__ out)         // (8192, 512)
{
  const int lane = threadIdx.x & 31;
  const int wave = threadIdx.x >> 5;
  const int tile = blockIdx.x * 8 + wave;   // 0 .. 16383
  const int mt   = tile >> 5;               // row tile   0..511
  const int nt   = tile & 31;               // col tile   0..31

  const int r    = lane & 15;               // position within half-wave
  const int half = lane >> 4;               // 0: K slots {0,1}, 1: K slots {2,3}

  // Each lane streams 16B chunks of one x-row and one w-row.
  // A-frag (16x4, MxK): vreg0/vreg1 hold two consecutive K per lane-half.
  // B-frag (4x16, KxN): same slot layout with N = r, sourced from W_all row.
  const float* xrow = x + (size_t)(mt * 16 + r) * IN_F + 4 * half;
  const float* wrow = w + (size_t)(nt * 16 + r) * IN_F + 4 * half;

  v8f c = {};  // 16x16 f32 accumulator (8 VGPRs)

#pragma unroll 4
  for (int j = 0; j < 32; ++j) {
    // half=0 covers K = j*8 .. j*8+3, half=1 covers K = j*8+4 .. j*8+7.
    // A and B use the identical K->slot assignment, so accumulation is exact.
    const v4f xa = *(const v4f*)(xrow + j * 8);
    const v4f wb = *(const v4f*)(wrow + j * 8);

    v2f a0; a0.x = xa.x; a0.y = xa.y;
    v2f b0; b0.x = wb.x; b0.y = wb.y;
    c = __builtin_amdgcn_wmma_f32_16x16x4_f32(
        /*neg_a=*/false, a0, /*neg_b=*/false, b0,
        /*c_mod=*/(short)0, c, /*reuse_a=*/false, /*reuse_b=*/false);

    v2f a1; a1.x = xa.z; a1.y = xa.w;
    v2f b1; b1.x = wb.z; b1.y = wb.w;
    c = __builtin_amdgcn_wmma_f32_16x16x4_f32(
        /*neg_a=*/false, a1, /*neg_b=*/false, b1,
        /*c_mod=*/(short)0, c, /*reuse_a=*/false, /*reuse_b=*/false);
  }

  // D layout: vreg i, lane L -> M = i + 8*(L/16), N = L%16.
  const int ncol = nt * 16 + r;
  float* orow = out + (size_t)(mt * 16 + 8 * half) * OUT_ALL + ncol;
#pragma unroll
  for (int i = 0; i < 8; ++i) {
    orow[(size_t)i * OUT_ALL] = c[i] * 0.0078125f;  // * 1/128 (exact)
  }
}

extern "C" void kernel_launch(void* const* d_in, const int* in_sizes, int n_in,
                              void* d_out, int out_size, void* d_ws, size_t ws_size,
                              hipStream_t stream) {
  const float* x = (const float*)d_in[0];   // (8192, 256)
  // d_in[1] = adjacency: provably unused (softmax of row-constant == 1/128)
  const float* w = (const float*)d_in[2];   // (4, 128, 256) -> (512, 256)
  float* out = (float*)d_out;               // (8192, 512)

  // 512 row-tiles * 32 col-tiles = 16384 tiles; 8 waves per 256-thread block.
  const int tiles = (N_ROWS / 16) * (OUT_ALL / 16);
  const int blocks = tiles / 8;             // 2048
  gat_gemm_wmma_f32<<<blocks, 256, 0, stream>>>(x, w, out);
}